// CenterLossLayer_87522843560826
// MI455X (gfx1250) — compile-verified
//
#include <hip/hip_runtime.h>
#include <hip/hip_bf16.h>

#define ALPHA       0.5f
#define NUM_CLASSES 100000
#define FEAT_DIM    128
#define BATCH       131072

typedef __attribute__((ext_vector_type(2))) float v2f;
typedef __attribute__((ext_vector_type(8))) float v8f;

// ---------------------------------------------------------------------------
// Kernel 1: label histogram (counts[c] = #samples with label c)
// ---------------------------------------------------------------------------
__global__ void cl_hist_kernel(const int* __restrict__ labels,
                               unsigned* __restrict__ counts) {
    int i = blockIdx.x * blockDim.x + threadIdx.x;
    if (i < BATCH) {
        atomicAdd(&counts[labels[i]], 1u);
    }
}

// ---------------------------------------------------------------------------
// Kernel 2: new_centers = centers (float4 grid-stride copy; scatter happens
// later as atomic adds on top of this baseline)
// ---------------------------------------------------------------------------
__global__ void cl_copy_kernel(const float* __restrict__ centers,
                               float* __restrict__ new_centers) {
    const size_t total4 = (size_t)NUM_CLASSES * FEAT_DIM / 4;
    size_t stride = (size_t)gridDim.x * blockDim.x;
    for (size_t i = (size_t)blockIdx.x * blockDim.x + threadIdx.x;
         i < total4; i += stride) {
        ((float4*)new_centers)[i] = ((const float4*)centers)[i];
    }
}

// ---------------------------------------------------------------------------
// Kernel 3 (fused): per 16-sample tile per wave:
//   diff = f - c staged in LDS (padded row stride 132 to spread banks)
//   scatter atomicAdd(ALPHA * diff / (1 + count)) into new_centers
//   loss via WMMA: acc(16x16) = diff * diff^T over K=128, take diagonal
// ---------------------------------------------------------------------------
#define WAVES_PER_BLOCK 4
#define ROW_STRIDE      132   // 128 + 4 pad: (132*m)%64 distinct for m=0..15

__global__ void
__launch_bounds__(32 * WAVES_PER_BLOCK)
cl_main_kernel(const float* __restrict__ features,
               const int* __restrict__ labels,
               const float* __restrict__ centers,
               const unsigned* __restrict__ counts,
               float* __restrict__ out_loss,
               float* __restrict__ out_centers) {
    __shared__ float diff_s[WAVES_PER_BLOCK][16 * ROW_STRIDE];
    __shared__ float diag_s[WAVES_PER_BLOCK][16 * 16];

    const int lane = threadIdx.x & 31;
    const int wave = threadIdx.x >> 5;
    const int s0   = (blockIdx.x * WAVES_PER_BLOCK + wave) * 16;

    float* tile = &diff_s[wave][0];

    // ---- Stage 1: gather, diff, scatter-update of new_centers -------------
    // lane covers cols [4*lane, 4*lane+3] of every row -> fully coalesced.
    #pragma unroll 4
    for (int r = 0; r < 16; ++r) {
        const int    s   = s0 + r;
        const int    lab = labels[s];
        const float  scale = ALPHA / (1.0f + (float)counts[lab]);
        const size_t col = (size_t)lane * 4;

        const float4 f4 = *(const float4*)(features + (size_t)s   * FEAT_DIM + col);
        const float4 c4 = *(const float4*)(centers  + (size_t)lab * FEAT_DIM + col);

        float4 d;
        d.x = f4.x - c4.x;  d.y = f4.y - c4.y;
        d.z = f4.z - c4.z;  d.w = f4.w - c4.w;

        float* row = tile + r * ROW_STRIDE + lane * 4;
        row[0] = d.x; row[1] = d.y; row[2] = d.z; row[3] = d.w;

        float* dst = out_centers + (size_t)lab * FEAT_DIM + col;
        atomicAdd(dst + 0, d.x * scale);
        atomicAdd(dst + 1, d.y * scale);
        atomicAdd(dst + 2, d.z * scale);
        atomicAdd(dst + 3, d.w * scale);
    }
    __syncthreads();

    // ---- Stage 2: acc(16x16) += diff_tile(16x4) * diff_tile^T(4x16) -------
    // A-fragment layout (32-bit, 16x4): lane L holds row m=L%16,
    // VGPR0 = K = k0 + 2*(L/16), VGPR1 = K+1. For B = A^T the striped
    // B-fragment carries identical values, so one fragment feeds both.
    const int m    = lane & 15;
    const int koff = (lane >> 4) * 2;
    const float* arow = tile + m * ROW_STRIDE + koff;

    v8f acc = {};
    #pragma unroll
    for (int k0 = 0; k0 < FEAT_DIM; k0 += 4) {
        v2f a;
        a.x = arow[k0 + 0];
        a.y = arow[k0 + 1];
        acc = __builtin_amdgcn_wmma_f32_16x16x4_f32(
            /*neg_a=*/false, a, /*neg_b=*/false, a,
            /*c_mod=*/(short)0, acc, /*reuse_a=*/false, /*reuse_b=*/false);
    }

    // ---- Stage 3: extract diagonal = per-sample loss ----------------------
    // C/D layout: VGPR r, lanes 0-15 -> (M=r,   N=lane),
    //                     lanes 16-31 -> (M=r+8, N=lane-16)
    {
        float* dg = &diag_s[wave][0];
        const int rowbase = (lane >> 4) * 8;
        const int n = lane & 15;
        #pragma unroll
        for (int r = 0; r < 8; ++r) {
            dg[(rowbase + r) * 16 + n] = acc[r];
        }
    }
    __syncthreads();
    if (lane < 16) {
        out_loss[s0 + lane] = diag_s[wave][lane * 16 + lane];
    }
}

// ---------------------------------------------------------------------------
// Launch
// ---------------------------------------------------------------------------
extern "C" void kernel_launch(void* const* d_in, const int* in_sizes, int n_in,
                              void* d_out, int out_size, void* d_ws, size_t ws_size,
                              hipStream_t stream) {
    const float* features = (const float*)d_in[0];  // [B, 128]
    const int*   labels   = (const int*)d_in[1];    // [B]
    const float* centers  = (const float*)d_in[2];  // [C, 128]

    float* out_loss    = (float*)d_out;             // [B] (result, flattened)
    float* out_centers = out_loss + BATCH;          // [C, 128]

    unsigned* counts = (unsigned*)d_ws;             // [C]

    hipMemsetAsync(counts, 0, (size_t)NUM_CLASSES * sizeof(unsigned), stream);

    cl_hist_kernel<<<(BATCH + 255) / 256, 256, 0, stream>>>(labels, counts);

    cl_copy_kernel<<<1024, 256, 0, stream>>>(centers, out_centers);

    const int tiles  = BATCH / 16;                      // 8192 waves
    const int blocks = tiles / WAVES_PER_BLOCK;         // 2048 blocks
    cl_main_kernel<<<blocks, 32 * WAVES_PER_BLOCK, 0, stream>>>(
        features, labels, centers, counts, out_loss, out_centers);
}